// GATModel_33225867002444
// MI455X (gfx1250) — compile-verified
//
#include <hip/hip_runtime.h>

typedef float v2f __attribute__((ext_vector_type(2)));
typedef float v8f __attribute__((ext_vector_type(8)));

#define HC   128   // H*C
#define H    4
#define C    32
#define GRPS 64
#define OUTD 2

// ---------- monotone float<->int encoding for atomic max ----------
__device__ __forceinline__ int enc_f(float f) {
    int i = __float_as_int(f);
    return i >= 0 ? i : (i ^ 0x7fffffff);
}
__device__ __forceinline__ float dec_f(int i) {
    return __int_as_float(i >= 0 ? i : (i ^ 0x7fffffff));
}

// ---------- 1) xp = x @ W via V_WMMA_F32_16X16X4_F32 ----------
// block = 256 threads (8 waves). Each block: one 16-row tile of x (staged in LDS),
// each wave: one 16-col tile of W. K-loop: 128/4 = 32 WMMAs.
__global__ void __launch_bounds__(256)
gemm_xp_kernel(const float* __restrict__ x, const float* __restrict__ W,
               float* __restrict__ xp, int N) {
    __shared__ __align__(16) float As[16 * 128];
    const int lane = threadIdx.x & 31;
    const int wave = threadIdx.x >> 5;
    const int row0 = blockIdx.x << 4;
    const int col0 = wave << 4;
    const int half = lane >> 4;   // 0: K=0,1  1: K=2,3 (per ISA 16x4 fp32 A layout)
    const int l16  = lane & 15;

    // cooperative A-tile load: 16 rows x 128 floats = 512 float4
    for (int i = threadIdx.x; i < 512; i += 256) {
        int r  = i >> 5;
        int c4 = i & 31;
        int gr = row0 + r;
        float4 v = make_float4(0.f, 0.f, 0.f, 0.f);
        if (gr < N) v = ((const float4*)(x + (size_t)gr * 128))[c4];
        ((float4*)As)[i] = v;
    }
    __syncthreads();

    v8f acc = {0.f, 0.f, 0.f, 0.f, 0.f, 0.f, 0.f, 0.f};
#pragma unroll
    for (int kk = 0; kk < 128; kk += 4) {
        int ka = kk + half * 2;
        v2f a, b;
        a[0] = As[l16 * 128 + ka];
        a[1] = As[l16 * 128 + ka + 1];
        b[0] = W[(size_t)ka * HC + col0 + l16];
        b[1] = W[(size_t)(ka + 1) * HC + col0 + l16];
        acc = __builtin_amdgcn_wmma_f32_16x16x4_f32(
            false, a, false, b, (short)0, acc, false, false);
    }
    // D layout: VGPR r -> lanes 0-15: M=r, lanes 16-31: M=r+8
#pragma unroll
    for (int r = 0; r < 8; ++r) {
        int row = row0 + r + half * 8;
        if (row < N) xp[(size_t)row * HC + col0 + l16] = acc[r];
    }
}

// ---------- 2) per-(node, head) attention logits ----------
__global__ void logits_kernel(const float* __restrict__ xp,
                              const float* __restrict__ att_src,
                              const float* __restrict__ att_dst,
                              float* __restrict__ asrc, float* __restrict__ adst,
                              int N) {
    int t = blockIdx.x * blockDim.x + threadIdx.x;
    if (t >= N * H) return;
    int h = t & (H - 1);
    const float4* v  = (const float4*)(xp + (size_t)t * C);
    const float4* as = (const float4*)(att_src + h * C);
    const float4* ad = (const float4*)(att_dst + h * C);
    float s = 0.f, d = 0.f;
#pragma unroll
    for (int i = 0; i < C / 4; ++i) {
        float4 vv = v[i], a = as[i], b = ad[i];
        s += vv.x * a.x + vv.y * a.y + vv.z * a.z + vv.w * a.w;
        d += vv.x * b.x + vv.y * b.y + vv.z * b.z + vv.w * b.w;
    }
    asrc[t] = s;
    adst[t] = d;
}

// ---------- 0) init workspace ----------
__global__ void init_kernel(float* outf, float* denom, int* emax,
                            float* pooled, float* counts, int N) {
    int t = blockIdx.x * blockDim.x + threadIdx.x;
    if (t < N * HC) outf[t] = 0.f;
    if (t < N * H) { denom[t] = 0.f; emax[t] = (int)0x80000000; }
    if (t < GRPS * HC) pooled[t] = 0.f;
    if (t < GRPS) counts[t] = 0.f;
}

// ---------- 3) per-(edge, head) logit + leaky relu + segment max ----------
__global__ void edge_max_kernel(const int* __restrict__ esrc, const int* __restrict__ edst,
                                int E, int ET,
                                const float* __restrict__ asrc, const float* __restrict__ adst,
                                float* __restrict__ ebuf, int* __restrict__ emax) {
    int t = blockIdx.x * blockDim.x + threadIdx.x;
    if (t >= ET * H) return;
    int e = t >> 2, h = t & (H - 1);
    int s, d;
    if (e < E) { s = esrc[e]; d = edst[e]; } else { s = d = e - E; }
    float v = asrc[s * H + h] + adst[d * H + h];
    v = v >= 0.f ? v : 0.2f * v;          // leaky relu, slope 0.2
    ebuf[t] = v;
    atomicMax(&emax[d * H + h], enc_f(v));
}

// ---------- 4) exp(e - max) + segment sum ----------
__global__ void edge_exp_kernel(const int* __restrict__ esrc, const int* __restrict__ edst,
                                int E, int ET,
                                const int* __restrict__ emax,
                                float* __restrict__ ebuf, float* __restrict__ denom) {
    int t = blockIdx.x * blockDim.x + threadIdx.x;
    if (t >= ET * H) return;
    int e = t >> 2, h = t & (H - 1);
    int d = (e < E) ? edst[e] : (e - E);
    float m  = dec_f(emax[d * H + h]);
    float ee = __expf(ebuf[t] - m);
    ebuf[t] = ee;
    atomicAdd(&denom[d * H + h], ee);
}

// ---------- 5) wave-per-edge weighted scatter-add ----------
// lane covers features [lane*4, lane*4+4) -> all within head h = lane>>3
__global__ void __launch_bounds__(256)
aggregate_kernel(const int* __restrict__ esrc, const int* __restrict__ edst,
                 int E, int ET,
                 const float* __restrict__ xp, const float* __restrict__ ebuf,
                 const float* __restrict__ denom, float* __restrict__ outf) {
    int wid  = (blockIdx.x * blockDim.x + threadIdx.x) >> 5;
    int lane = threadIdx.x & 31;
    if (wid >= ET) return;
    int s, d;
    if (wid < E) { s = esrc[wid]; d = edst[wid]; } else { s = d = wid - E; }
    int h = lane >> 3;
    float den   = denom[d * H + h];
    float alpha = ebuf[wid * H + h] / fmaxf(den, 1e-16f);
    float4 v = ((const float4*)(xp + (size_t)s * HC))[lane];
    float* o = outf + (size_t)d * HC + lane * 4;
    atomicAdd(o + 0, alpha * v.x);
    atomicAdd(o + 1, alpha * v.y);
    atomicAdd(o + 2, alpha * v.z);
    atomicAdd(o + 3, alpha * v.w);
}

// ---------- 6) bias + relu + per-graph mean-pool (sum + count) ----------
__global__ void pool_kernel(const float* __restrict__ outf, const float* __restrict__ bias,
                            const int* __restrict__ batch,
                            float* __restrict__ pooled, float* __restrict__ counts, int N) {
    int t = blockIdx.x * blockDim.x + threadIdx.x;
    if (t >= N * (HC / 4)) return;
    int n = t >> 5, q = t & 31;
    float4 o = ((const float4*)(outf + (size_t)n * HC))[q];
    float4 b = ((const float4*)bias)[q];
    o.x = fmaxf(o.x + b.x, 0.f);
    o.y = fmaxf(o.y + b.y, 0.f);
    o.z = fmaxf(o.z + b.z, 0.f);
    o.w = fmaxf(o.w + b.w, 0.f);
    int g = batch[n];
    float* p = pooled + (size_t)g * HC + q * 4;
    atomicAdd(p + 0, o.x);
    atomicAdd(p + 1, o.y);
    atomicAdd(p + 2, o.z);
    atomicAdd(p + 3, o.w);
    if (q == 0) atomicAdd(&counts[g], 1.0f);
}

// ---------- 7) mean + final linear [64,128]@[128,2] + bias ----------
__global__ void final_kernel(const float* __restrict__ pooled, const float* __restrict__ counts,
                             const float* __restrict__ W_lin, const float* __restrict__ b_lin,
                             float* __restrict__ out) {
    int t = threadIdx.x;
    if (t >= GRPS * OUTD) return;
    int g = t >> 1, o = t & 1;
    float inv = 1.0f / fmaxf(counts[g], 1.0f);
    float acc = 0.f;
#pragma unroll 8
    for (int k = 0; k < HC; ++k)
        acc += pooled[(size_t)g * HC + k] * W_lin[k * OUTD + o];
    out[t] = acc * inv + b_lin[o] * 1.0f;  // mean then affine (inv distributes over sum)
}

extern "C" void kernel_launch(void* const* d_in, const int* in_sizes, int n_in,
                              void* d_out, int out_size, void* d_ws, size_t ws_size,
                              hipStream_t stream) {
    const float* x       = (const float*)d_in[0];
    const int*   edge    = (const int*)d_in[1];
    const int*   batch   = (const int*)d_in[2];
    const float* W       = (const float*)d_in[3];
    const float* att_src = (const float*)d_in[4];
    const float* att_dst = (const float*)d_in[5];
    const float* bias    = (const float*)d_in[6];
    const float* W_lin   = (const float*)d_in[7];
    const float* b_lin   = (const float*)d_in[8];

    const int N  = in_sizes[0] / HC;   // F_IN == HC == 128
    const int E  = in_sizes[1] / 2;
    const int ET = E + N;              // edges + self loops

    // workspace carve-out (256B aligned slots)
    char* wsp = (char*)d_ws;
    auto carve = [&](size_t bytes) {
        char* p = wsp;
        wsp += (bytes + 255) & ~((size_t)255);
        return p;
    };
    float* xp     = (float*)carve((size_t)N * HC * 4);
    float* outf   = (float*)carve((size_t)N * HC * 4);
    float* ebuf   = (float*)carve((size_t)ET * H * 4);
    float* asrc   = (float*)carve((size_t)N * H * 4);
    float* adst   = (float*)carve((size_t)N * H * 4);
    int*   emax   = (int*)  carve((size_t)N * H * 4);
    float* denom  = (float*)carve((size_t)N * H * 4);
    float* pooled = (float*)carve((size_t)GRPS * HC * 4);
    float* counts = (float*)carve((size_t)GRPS * 4);

    const int* esrc = edge;
    const int* edst = edge + E;

    // 0) init
    {
        int total = N * HC;
        init_kernel<<<(total + 255) / 256, 256, 0, stream>>>(outf, denom, emax, pooled, counts, N);
    }
    // 1) xp = x @ W (WMMA)
    gemm_xp_kernel<<<(N + 15) / 16, 256, 0, stream>>>(x, W, xp, N);
    // 2) attention logits
    logits_kernel<<<(N * H + 255) / 256, 256, 0, stream>>>(xp, att_src, att_dst, asrc, adst, N);
    // 3) edge logits + segment max
    edge_max_kernel<<<(ET * H + 255) / 256, 256, 0, stream>>>(esrc, edst, E, ET, asrc, adst, ebuf, emax);
    // 4) exp + segment sum
    edge_exp_kernel<<<(ET * H + 255) / 256, 256, 0, stream>>>(esrc, edst, E, ET, emax, ebuf, denom);
    // 5) aggregation (one wave per edge)
    aggregate_kernel<<<(ET + 7) / 8, 256, 0, stream>>>(esrc, edst, E, ET, xp, ebuf, denom, outf);
    // 6) bias + relu + pool
    pool_kernel<<<(N * (HC / 4) + 255) / 256, 256, 0, stream>>>(outf, bias, batch, pooled, counts, N);
    // 7) mean + final linear
    final_kernel<<<1, 128, 0, stream>>>(pooled, counts, W_lin, b_lin, (float*)d_out);
}